// RelativePositionalEncoding_25872882991585
// MI455X (gfx1250) — compile-verified
//
#include <hip/hip_runtime.h>
#include <math.h>

// Problem constants (fixed by the reference setup)
#define RPE_B 32
#define RPE_S 4096
#define RPE_D 256
#define RPE_HALF_D 128
#define RPE_MAX_LEN 5000

typedef float v4f __attribute__((ext_vector_type(4)));

// -ln(10000) = -9.210340371976184
#define NEG_LN_10000 (-9.210340371976184f)

// ---------------------------------------------------------------------------
// Kernel 1: build the PE table into workspace: pe[p][2k] = sin(p*div_k),
// pe[p][2k+1] = cos(p*div_k), div_k = exp(2k * -ln(10000)/256).
// 5000*128 = 640K sincos pairs -> ~1-2us, amortizes the transcendental cost
// out of the 33.5M-element hot loop. Table (5.12 MB) stays resident in the
// 192 MB L2 for the main kernel's row gathers.
// ---------------------------------------------------------------------------
__global__ __launch_bounds__(256) void rpe_pe_build_kernel(float* __restrict__ pe) {
    int idx = blockIdx.x * 256 + threadIdx.x;
    if (idx >= RPE_MAX_LEN * RPE_HALF_D) return;
    int p = idx >> 7;        // position 0..4999
    int k = idx & 127;       // frequency index 0..127
    float div = expf((float)(2 * k) * (NEG_LN_10000 / (float)RPE_D));
    float ang = (float)p * div;
    float s, c;
    sincosf(ang, &s, &c);    // accurate library version (large-arg reduction)
    pe[p * RPE_D + 2 * k]     = s;
    pe[p * RPE_D + 2 * k + 1] = c;
}

// ---------------------------------------------------------------------------
// Kernel 2: out = nodes + (s <= n_b ? pe[(s - n_b) mod 5000] : 0)
// 256 threads (8 wave32s) per block, one float4 per thread -> 4 rows/block.
// 4096 rows per batch is divisible by 4 -> batch index b is block-uniform
// (scalar load of num_nodes[b]); each wave32 covers half a row -> the mask
// branch is wave-uniform (no divergence).
// nodes/out are 128 MiB streams touched once -> non-temporal (TH=NT) b128
// loads/stores so they don't thrash the L2-resident PE table; PE loads use
// default RT policy so the table stays hot in L2.
// ---------------------------------------------------------------------------
__global__ __launch_bounds__(256) void rpe_add_kernel(
    const float* __restrict__ nodes,
    const int*   __restrict__ num_nodes,
    const float* __restrict__ pe,
    float*       __restrict__ out)
{
    const int tid       = threadIdx.x;
    const int rowInBlk  = tid >> 6;          // 64 float4 per 256-float row
    const int col4      = tid & 63;
    const int row       = blockIdx.x * 4 + rowInBlk;
    const int b         = row >> 12;         // row / 4096
    const int s         = row & (RPE_S - 1); // row % 4096
    const int n         = num_nodes[b];      // block-uniform -> s_load

    const size_t base = (size_t)row * RPE_D + (size_t)col4 * 4;

    v4f v = __builtin_nontemporal_load((const v4f*)(nodes + base));

    if (s <= n) {  // wave-uniform branch
        int idx = s - n;                     // in [-4095, 0] when masked
        if (idx < 0) idx += RPE_MAX_LEN;     // Python % semantics
        const v4f a = *(const v4f*)(pe + (size_t)idx * RPE_D + (size_t)col4 * 4);
        v += a;
    }

    __builtin_nontemporal_store(v, (v4f*)(out + base));
}

// ---------------------------------------------------------------------------
// Fallback: compute sin/cos inline if the workspace can't hold the PE table.
// Deterministic host-side selection based on ws_size (stable across calls,
// graph-capture safe).
// ---------------------------------------------------------------------------
__global__ __launch_bounds__(256) void rpe_add_inline_kernel(
    const float* __restrict__ nodes,
    const int*   __restrict__ num_nodes,
    float*       __restrict__ out)
{
    const int tid      = threadIdx.x;
    const int rowInBlk = tid >> 6;
    const int col4     = tid & 63;
    const int row      = blockIdx.x * 4 + rowInBlk;
    const int b        = row >> 12;
    const int s        = row & (RPE_S - 1);
    const int n        = num_nodes[b];

    const size_t base = (size_t)row * RPE_D + (size_t)col4 * 4;
    v4f v = __builtin_nontemporal_load((const v4f*)(nodes + base));

    if (s <= n) {
        int idx = s - n;
        if (idx < 0) idx += RPE_MAX_LEN;
        const float fp = (float)idx;
        #pragma unroll
        for (int j = 0; j < 2; ++j) {
            int k = col4 * 2 + j;            // frequency index for float pair
            float div = expf((float)(2 * k) * (NEG_LN_10000 / (float)RPE_D));
            float sn, cs;
            sincosf(fp * div, &sn, &cs);
            v[2 * j]     += sn;
            v[2 * j + 1] += cs;
        }
    }

    __builtin_nontemporal_store(v, (v4f*)(out + base));
}

extern "C" void kernel_launch(void* const* d_in, const int* in_sizes, int n_in,
                              void* d_out, int out_size, void* d_ws, size_t ws_size,
                              hipStream_t stream) {
    const float* nodes     = (const float*)d_in[0];
    const int*   num_nodes = (const int*)d_in[1];
    float*       out       = (float*)d_out;

    const size_t peBytes = (size_t)RPE_MAX_LEN * RPE_D * sizeof(float); // 5.12 MB
    const int    nRows   = RPE_B * RPE_S;                               // 131072
    const int    nBlocks = nRows / 4;                                   // 32768

    if (ws_size >= peBytes) {
        float* pe = (float*)d_ws;
        const int peThreads = RPE_MAX_LEN * RPE_HALF_D;                 // 640000
        rpe_pe_build_kernel<<<(peThreads + 255) / 256, 256, 0, stream>>>(pe);
        rpe_add_kernel<<<nBlocks, 256, 0, stream>>>(nodes, num_nodes, pe, out);
    } else {
        rpe_add_inline_kernel<<<nBlocks, 256, 0, stream>>>(nodes, num_nodes, out);
    }
}